// BdcPool_7447473291738
// MI455X (gfx1250) — compile-verified
//
#include <hip/hip_runtime.h>
#include <math.h>

// ---------------------------------------------------------------------------
// Brownian-distance-covariance pooling (BdcPool) for MI455X / gfx1250.
//   B=64, DIM=640, M=H*W=100, out per batch = DIM*(DIM+1)/2 = 205120 f32.
// Strategy: bandwidth-bound problem -> recompute Gram tiles with
// V_WMMA_F32_16X16X4_F32 in two passes instead of materializing the 105 MB
// dcov tensor. Deterministic (no FP atomics): per-tile row partials in ws.
// ---------------------------------------------------------------------------

typedef float v2f __attribute__((ext_vector_type(2)));
typedef float v8f __attribute__((ext_vector_type(8)));

#define BATCH 64
#define DIM   640
#define KM    100          // H*W
#define NTRI  205120       // DIM*(DIM+1)/2
#define TILE  64           // 64x64 C-tile per block
#define NT    10           // DIM / TILE
#define LDA   104          // LDS row stride (floats), padded vs 100

// ---------------------------------------------------------------------------
// Kernel 0: diag[b,i] = sum_m x[b,i,m]^2   (row index bi = b*DIM + i)
// ---------------------------------------------------------------------------
__global__ void k_diag(const float* __restrict__ x, float* __restrict__ diag) {
    int idx = blockIdx.x * blockDim.x + threadIdx.x;
    if (idx >= BATCH * DIM) return;
    const float* row = x + (size_t)idx * KM;
    float s = 0.f;
#pragma unroll 10
    for (int m = 0; m < KM; ++m) { float v = row[m]; s = fmaf(v, v, s); }
    diag[idx] = s;
}

// ---------------------------------------------------------------------------
// Shared tile machinery: one 256-thread block computes a 64x64 tile of
// G = X X^T for one batch via v_wmma_f32_16x16x4_f32.
// Wave w (0..7): wy = w&3 selects M-chunk (16 rows), wx = w>>2 selects a pair
// of N-chunks (2 x 16 cols) -> two v8f accumulators per wave.
// A-frag (16x4 f32): lane L holds M=L%16; VGPR v holds K = k0 + v + 2*(L>>4).
// B-frag (4x16 f32): lane L holds N=L%16; same K striping (B[k][n]=X[j0+n][k]).
// ---------------------------------------------------------------------------
__device__ __forceinline__ void gram_tile_compute(
    const float* __restrict__ Xa, const float* __restrict__ Xb,
    int wy, int wx, int g, int l16, v8f& c0, v8f& c1)
{
    const float* Ar = &Xa[(16 * wy + l16) * LDA];
    const float* B0 = &Xb[(16 * (2 * wx + 0) + l16) * LDA];
    const float* B1 = &Xb[(16 * (2 * wx + 1) + l16) * LDA];
    for (int k0 = 0; k0 < KM; k0 += 4) {
        const int kk = k0 + 2 * g;
        v2f a  = { Ar[kk], Ar[kk + 1] };
        v2f b0 = { B0[kk], B0[kk + 1] };
        v2f b1 = { B1[kk], B1[kk + 1] };
        c0 = __builtin_amdgcn_wmma_f32_16x16x4_f32(false, a, false, b0,
                                                   (short)0, c0, false, false);
        c1 = __builtin_amdgcn_wmma_f32_16x16x4_f32(false, a, false, b1,
                                                   (short)0, c1, false, false);
    }
}

// stage 64x100 X rows into LDS (stride LDA) -- 25 iters/thread, coalesced
__device__ __forceinline__ void stage_tiles(
    const float* __restrict__ x, int b, int i0, int j0, int tid,
    float* __restrict__ Xa, float* __restrict__ Xb)
{
    const float* xa = x + ((size_t)b * DIM + i0) * KM;
    const float* xb = x + ((size_t)b * DIM + j0) * KM;
    for (int idx = tid; idx < TILE * KM; idx += 256) {
        int r = idx / KM, c = idx - r * KM;
        Xa[r * LDA + c] = xa[r * KM + c];
        Xb[r * LDA + c] = xb[r * KM + c];
    }
}

// ---------------------------------------------------------------------------
// Pass A: full 10x10 tile grid. Compute dcov elements, reduce rows across
// lanes via LDS, emit per-tile row partial sums:
//   rowpart[(b*DIM + i) * NT + jt] = sum_{j in tile jt} dcov[b,i,j]
// ---------------------------------------------------------------------------
__global__ __launch_bounds__(256)
void k_gram_rowpart(const float* __restrict__ x, const float* __restrict__ diag,
                    const float* __restrict__ temp, float* __restrict__ rowpart)
{
    __shared__ float Xa[TILE * LDA];
    __shared__ float Xb[TILE * LDA];
    __shared__ float dA[TILE];
    __shared__ float dB[TILE];
    __shared__ float rowAcc[TILE * 33];   // 64 rows x 32 slots, padded

    const int jt = blockIdx.x, it = blockIdx.y, b = blockIdx.z;
    const int i0 = it * TILE, j0 = jt * TILE;
    const int tid = threadIdx.x;

    stage_tiles(x, b, i0, j0, tid, Xa, Xb);
    {
        const int t = tid & 63, sel = tid >> 6;
        const size_t bD = (size_t)b * DIM;
        if (sel == 0)      dA[t] = diag[bD + i0 + t];
        else if (sel == 1) dB[t] = diag[bD + j0 + t];
    }
    __syncthreads();

    const int wave = tid >> 5, lane = tid & 31;
    const int wy = wave & 3, wx = wave >> 2;
    const int g = lane >> 4, l16 = lane & 15;

    v8f c0 = {}; v8f c1 = {};
    gram_tile_compute(Xa, Xb, wy, wx, g, l16, c0, c1);

    const float expT = expf(temp[0]);
    const int N0 = 16 * (2 * wx + 0) + l16;
    const int N1 = 16 * (2 * wx + 1) + l16;
    const float dj0 = dB[N0], dj1 = dB[N1];

#pragma unroll
    for (int r = 0; r < 8; ++r) {
        const int Ml = 16 * wy + r + 8 * g;
        const float di = dA[Ml];
        float e0 = di + dj0 - 2.f * c0[r];
        float e1 = di + dj1 - 2.f * c1[r];
        e0 = sqrtf(fmaf(expT, fmaxf(e0, 0.f), 1e-5f));
        e1 = sqrtf(fmaf(expT, fmaxf(e1, 0.f), 1e-5f));
        rowAcc[Ml * 33 + (wx * 16 + l16)] = e0 + e1;   // unique slot per thread
    }
    __syncthreads();

    if (tid < TILE) {
        float s = 0.f;
#pragma unroll 8
        for (int k = 0; k < 32; ++k) s += rowAcc[tid * 33 + k];
        rowpart[((size_t)b * DIM + i0 + tid) * NT + jt] = s;
    }
}

// ---------------------------------------------------------------------------
// rowsum[b,i] = sum_jt rowpart;  total[b] = sum_i rowsum[b,i]  (deterministic)
// ---------------------------------------------------------------------------
__global__ void k_rowsum_total(const float* __restrict__ rowpart,
                               float* __restrict__ rowsum,
                               float* __restrict__ total)
{
    __shared__ float sm[256];
    const int b = blockIdx.x, tid = threadIdx.x;
    float acc = 0.f;
    for (int i = tid; i < DIM; i += 256) {
        const float* p = rowpart + ((size_t)b * DIM + i) * NT;
        float s = 0.f;
#pragma unroll
        for (int jt = 0; jt < NT; ++jt) s += p[jt];
        rowsum[(size_t)b * DIM + i] = s;
        acc += s;
    }
    sm[tid] = acc;
    __syncthreads();
    for (int off = 128; off > 0; off >>= 1) {
        if (tid < off) sm[tid] += sm[tid + off];
        __syncthreads();
    }
    if (tid == 0) total[b] = sm[0];
}

// ---------------------------------------------------------------------------
// Pass B: upper-triangular tiles only (jt >= it). Recompute Gram tile with
// WMMA, center with staged rowmeans + totalmean, scatter triu output:
//   out[b*NTRI + i*(2*DIM - i + 1)/2 + (j - i)]   for j >= i
// ---------------------------------------------------------------------------
__global__ __launch_bounds__(256)
void k_gram_out(const float* __restrict__ x, const float* __restrict__ diag,
                const float* __restrict__ temp, const float* __restrict__ rowsum,
                const float* __restrict__ total, float* __restrict__ out)
{
    const int jt = blockIdx.x, it = blockIdx.y, b = blockIdx.z;
    if (jt < it) return;                       // uniform: EXEC stays all-ones

    __shared__ float Xa[TILE * LDA];
    __shared__ float Xb[TILE * LDA];
    __shared__ float dA[TILE];
    __shared__ float dB[TILE];
    __shared__ float rmA[TILE];                // rowmean slice (already /DIM)
    __shared__ float rmB[TILE];

    const int i0 = it * TILE, j0 = jt * TILE;
    const int tid = threadIdx.x;
    const float invD = 1.0f / (float)DIM;

    stage_tiles(x, b, i0, j0, tid, Xa, Xb);
    {
        const int t = tid & 63, sel = tid >> 6;
        const size_t bD = (size_t)b * DIM;
        if (sel == 0)      dA[t]  = diag[bD + i0 + t];
        else if (sel == 1) dB[t]  = diag[bD + j0 + t];
        else if (sel == 2) rmA[t] = rowsum[bD + i0 + t] * invD;
        else               rmB[t] = rowsum[bD + j0 + t] * invD;
    }
    __syncthreads();

    const int wave = tid >> 5, lane = tid & 31;
    const int wy = wave & 3, wx = wave >> 2;
    const int g = lane >> 4, l16 = lane & 15;

    v8f c0 = {}; v8f c1 = {};
    gram_tile_compute(Xa, Xb, wy, wx, g, l16, c0, c1);

    const float expT  = expf(temp[0]);
    const float tmean = total[b] * invD * invD;
    float* ob = out + (size_t)b * NTRI;

#pragma unroll
    for (int half = 0; half < 2; ++half) {
        const int nc = 2 * wx + half;
        const int N  = 16 * nc + l16;
        const int j  = j0 + N;
        const float dj = dB[N], rj = rmB[N];
        const v8f cc = half ? c1 : c0;
#pragma unroll
        for (int r = 0; r < 8; ++r) {
            const int Ml = 16 * wy + r + 8 * g;
            const int i  = i0 + Ml;
            if (j >= i) {
                float e = dA[Ml] + dj - 2.f * cc[r];
                e = sqrtf(fmaf(expT, fmaxf(e, 0.f), 1e-5f));
                e = e - rmA[Ml] - rj + tmean;
                const size_t tri = ((size_t)i * (size_t)(2 * DIM - i + 1)) >> 1;
                ob[tri + (size_t)(j - i)] = e;
            }
        }
    }
}

// ---------------------------------------------------------------------------
// Workspace layout (floats): diag[B*DIM] | rowpart[B*DIM*NT] | rowsum[B*DIM]
//                            | total[B]        -> ~1.97 MB total
// ---------------------------------------------------------------------------
extern "C" void kernel_launch(void* const* d_in, const int* in_sizes, int n_in,
                              void* d_out, int out_size, void* d_ws, size_t ws_size,
                              hipStream_t stream)
{
    const float* x    = (const float*)d_in[0];
    const float* temp = (const float*)d_in[1];
    float* out = (float*)d_out;

    float* ws      = (float*)d_ws;
    float* diag    = ws;                                   // B*DIM
    float* rowpart = diag + (size_t)BATCH * DIM;           // B*DIM*NT
    float* rowsum  = rowpart + (size_t)BATCH * DIM * NT;   // B*DIM
    float* total   = rowsum + (size_t)BATCH * DIM;         // B

    k_diag<<<(BATCH * DIM + 255) / 256, 256, 0, stream>>>(x, diag);

    dim3 tgrid(NT, NT, BATCH);
    k_gram_rowpart<<<tgrid, 256, 0, stream>>>(x, diag, temp, rowpart);

    k_rowsum_total<<<BATCH, 256, 0, stream>>>(rowpart, rowsum, total);

    k_gram_out<<<tgrid, 256, 0, stream>>>(x, diag, temp, rowsum, total, out);
}